// PGExplainer_2688649527838
// MI455X (gfx1250) — compile-verified
//
#include <hip/hip_runtime.h>

typedef float v2f __attribute__((ext_vector_type(2)));
typedef float v8f __attribute__((ext_vector_type(8)));

#define D_EMB            128
#define HID              64
#define WAVES_PER_BLOCK  8
#define EDGES_PER_WAVE   16
#define EDGES_PER_BLOCK  (WAVES_PER_BLOCK * EDGES_PER_WAVE)   // 128
#define KSTEPS           64                                    // 256 / 4

// LDS holds W1 pre-swizzled into WMMA B-fragments:
// frag index [s=0..63][t=0..3][lane=0..31][j=0..1] (float each) = 16384 floats = 64 KB
#define SB_ELEMS (KSTEPS * 4 * 32 * 2)

__global__ __launch_bounds__(256) void pgx_edge_mlp(
    const float* __restrict__ src_embs,   // [N, 128]
    const float* __restrict__ dst_embs,   // [N, 128]
    const int*   __restrict__ src_ids,    // [E]
    const int*   __restrict__ dst_ids,    // [E]
    const float* __restrict__ W1,         // [256, 64] row-major
    const float* __restrict__ b1,         // [64]
    const float* __restrict__ W2,         // [64]
    const float* __restrict__ b2,         // [1]
    float*       __restrict__ out,        // [E]
    int nEdges)
{
    __shared__ float sB[SB_ELEMS];

    const int tid = threadIdx.x;

    // ---- Stage W1 into LDS in B-fragment layout (one-time 64 KB per block) ----
    // idx = ((s*4 + t)*32 + l)*2 + j  maps to  W1[k][n],
    //   k = s*4 + (l>=16 ? 2 : 0) + j ,  n = t*16 + (l & 15)
    for (int idx = tid; idx < SB_ELEMS; idx += 256) {
        int j = idx & 1;
        int l = (idx >> 1) & 31;
        int t = (idx >> 6) & 3;
        int s = idx >> 8;
        int k = s * 4 + ((l >> 4) << 1) + j;
        int n = t * 16 + (l & 15);
        sB[idx] = W1[k * HID + n];
    }
    __syncthreads();

    const int wave = tid >> 5;
    const int lane = tid & 31;
    const int base = (blockIdx.x * WAVES_PER_BLOCK + wave) * EDGES_PER_WAVE;
    if (base >= nEdges) return;   // E % 16 == 0, so active tiles are always full

    const int m    = lane & 15;          // edge within tile / N within tile
    const int koff = (lane >> 4) << 1;   // 0 for lanes 0-15, 2 for lanes 16-31

    // Per-lane gather pointers for the A operand (each lane owns edge `m`)
    const int sid = src_ids[base + m];
    const int did = dst_ids[base + m];
    const float* srow = src_embs + (size_t)sid * D_EMB;
    const float* drow = dst_embs + (size_t)did * D_EMB;

    v8f acc[4];
    #pragma unroll
    for (int t = 0; t < 4; ++t) acc[t] = (v8f){};

    const v2f* frag = (const v2f*)sB;    // one v2f per (kstep*4+t, lane)

    // ---- Layer 1: [16x256] @ [256x64] via V_WMMA_F32_16X16X4_F32 ----
    #pragma unroll 1
    for (int hsel = 0; hsel < 2; ++hsel) {          // 0: src half, 1: dst half of concat
        const float* row = hsel ? drow : srow;
        #pragma unroll
        for (int s = 0; s < KSTEPS / 2; ++s) {
            v2f a = *(const v2f*)(row + s * 4 + koff);          // 8B gather, L2-resident
            const int fs = (hsel * (KSTEPS / 2) + s) * 4;
            #pragma unroll
            for (int t = 0; t < 4; ++t) {
                v2f bfr = frag[(fs + t) * 32 + lane];           // conflict-free ds_load_b64
                acc[t] = __builtin_amdgcn_wmma_f32_16x16x4_f32(
                    false, a, false, bfr, (short)0, acc[t], false, false);
            }
        }
    }

    // ---- bias + ReLU + Layer 2 (h @ W2[64x1] + b2) ----
    float bias1[4], w2v[4];
    #pragma unroll
    for (int t = 0; t < 4; ++t) {
        bias1[t] = b1[t * 16 + m];
        w2v[t]   = W2[t * 16 + m];
    }

    float partial[8];
    #pragma unroll
    for (int r = 0; r < 8; ++r) partial[r] = 0.0f;

    #pragma unroll
    for (int t = 0; t < 4; ++t) {
        #pragma unroll
        for (int r = 0; r < 8; ++r) {
            float h = acc[t][r] + bias1[t];   // element (M = r + 8*(lane>=16), N = t*16 + m)
            h = fmaxf(h, 0.0f);
            partial[r] += h * w2v[t];
        }
    }

    // Reduce over the 16 lanes of each half-wave (N dimension, 64 values total)
    #pragma unroll
    for (int r = 0; r < 8; ++r) {
        #pragma unroll
        for (int off = 8; off >= 1; off >>= 1)
            partial[r] += __shfl_xor(partial[r], off, 32);
    }

    const float bias2 = b2[0];
    if (m == 0) {
        const int mb = (lane >> 4) * 8;   // lane 0 -> edges 0..7, lane 16 -> edges 8..15
        #pragma unroll
        for (int r = 0; r < 8; ++r)
            out[base + mb + r] = partial[r] + bias2;
    }
}

extern "C" void kernel_launch(void* const* d_in, const int* in_sizes, int n_in,
                              void* d_out, int out_size, void* d_ws, size_t ws_size,
                              hipStream_t stream) {
    const float* src_embs = (const float*)d_in[0];
    const float* dst_embs = (const float*)d_in[1];
    const int*   src_ids  = (const int*)  d_in[2];
    const int*   dst_ids  = (const int*)  d_in[3];
    const float* W1       = (const float*)d_in[4];
    const float* b1       = (const float*)d_in[5];
    const float* W2       = (const float*)d_in[6];
    const float* b2       = (const float*)d_in[7];
    float*       out      = (float*)d_out;

    const int nEdges = in_sizes[2];
    const int blocks = (nEdges + EDGES_PER_BLOCK - 1) / EDGES_PER_BLOCK;
    pgx_edge_mlp<<<blocks, 256, 0, stream>>>(src_embs, dst_embs, src_ids, dst_ids,
                                             W1, b1, W2, b2, out, nEdges);
}